// GraphConv_1580547967222
// MI455X (gfx1250) — compile-verified
//
#include <hip/hip_runtime.h>
#include <hip/hip_bf16.h>

typedef __attribute__((ext_vector_type(2))) float v2f;
typedef __attribute__((ext_vector_type(8))) float v8f;

// ---------------------------------------------------------------------------
// Generic  out[M x N] = scale * (A[M x K] @ W[N x ldw]^T(+koff))  + epilogue
//   epilogue: (+ bias[col]) (+ b2[col]*rowsum[row]*scale) (relu)
// One 16x16 output tile per wave32 via V_WMMA_F32_16X16X4_F32, K stepped by 4.
// ISA layouts (CDNA5 ISA):
//   A 16x4 f32 : lane l (half=l>>4, r16=l&15): v0=A[r16, 2*half], v1=A[r16, 2*half+1]
//   B 4x16 f32 : lane l: v0=B[2*half, l&15],  v1=B[2*half+1, l&15]   (B = W^T)
//   C/D 16x16  : vgpr r: row = r + 8*half, col = l&15
// Tile assignment is wave-uniform -> EXEC stays all-ones for every WMMA.
// ---------------------------------------------------------------------------
__global__ __launch_bounds__(128) void wmma_gemm_wt(
    const float* __restrict__ A, const float* __restrict__ W,
    float* __restrict__ out, int M, int N, int K, int ldw, int koff,
    const float* __restrict__ bias,    // len N or nullptr
    const float* __restrict__ rowsum,  // len M or nullptr
    const float* __restrict__ b2,      // len N (used with rowsum)
    float scale, int do_relu)
{
  const int lane  = threadIdx.x & 31;
  const int wave  = threadIdx.x >> 5;
  const int wpb   = blockDim.x >> 5;          // waves per block
  const int tilesN = N >> 4;
  const int tile  = blockIdx.x * wpb + wave;  // wave-uniform
  if (tile >= (M >> 4) * tilesN) return;      // whole-wave exit, EXEC stays full
  const int tm = (tile / tilesN) << 4;
  const int tn = (tile % tilesN) << 4;

  const int half = lane >> 4;
  const int l16  = lane & 15;

  const float* __restrict__ Arow = A + (size_t)(tm + l16) * K;          // A row (M-major)
  const float* __restrict__ Wrow = W + (size_t)(tn + l16) * ldw + koff; // W row = B column

  v8f c = {};
  for (int k = 0; k < K; k += 4) {
    const int kk = k + 2 * half;                       // 8B-aligned (kk even)
    v2f a = *reinterpret_cast<const v2f*>(Arow + kk);  // global_load_b64
    v2f b = *reinterpret_cast<const v2f*>(Wrow + kk);
    c = __builtin_amdgcn_wmma_f32_16x16x4_f32(
        /*neg_a=*/false, a, /*neg_b=*/false, b,
        /*c_mod=*/(short)0, c, /*reuse_a=*/false, /*reuse_b=*/false);
  }

  const int col = tn + l16;
#pragma unroll
  for (int r = 0; r < 8; ++r) {
    const int row = tm + r + 8 * half;
    float v = c[r] * scale;
    if (bias)   v += bias[col];
    if (rowsum) v += b2[col] * rowsum[row] * scale;
    if (do_relu) v = fmaxf(v, 0.0f);
    out[(size_t)row * N + col] = v;
  }
}

// ---------------------------------------------------------------------------
// S[i,h]    = sum_j adj[i,j] * relu(pa[j,h] + pb[i,h] + b1[h])
// rowsum[i] = sum_j adj[i,j]
// One block per node i (H=256 threads, 8 waves); adj row staged in LDS so the
// per-j weight is a conflict-free LDS broadcast; pa rows stream from L2.
// ---------------------------------------------------------------------------
__global__ __launch_bounds__(256) void edge_aggregate(
    const float* __restrict__ pa, const float* __restrict__ pb,
    const float* __restrict__ b1, const float* __restrict__ adj,
    float* __restrict__ S, float* __restrict__ rowsum, int N, int H)
{
  __shared__ float s_adj[512];
  const int i = blockIdx.x;
  const int h = threadIdx.x;

  for (int j = h; j < N; j += blockDim.x)
    s_adj[j] = adj[(size_t)i * N + j];
  __syncthreads();

  const float cst = pb[(size_t)i * H + h] + b1[h];
  float acc = 0.0f, rs = 0.0f;
#pragma unroll 4
  for (int j = 0; j < N; ++j) {
    if ((j & 63) == 0 && j + 64 < N)
      __builtin_prefetch(pa + (size_t)(j + 64) * H + h, 0, 0); // global_prefetch
    const float w = s_adj[j];                                  // ds broadcast
    rs += w;
    acc = fmaf(w, fmaxf(pa[(size_t)j * H + h] + cst, 0.0f), acc);
  }
  S[(size_t)i * H + h] = acc;
  if (h == 0) rowsum[i] = rs;
}

// ---------------------------------------------------------------------------
extern "C" void kernel_launch(void* const* d_in, const int* in_sizes, int n_in,
                              void* d_out, int out_size, void* d_ws, size_t ws_size,
                              hipStream_t stream)
{
  const float* x   = (const float*)d_in[0];
  const float* adj = (const float*)d_in[1];
  const float* W1  = (const float*)d_in[2];
  const float* b1  = (const float*)d_in[3];
  const float* W2  = (const float*)d_in[4];
  const float* b2  = (const float*)d_in[5];
  const float* Wo1 = (const float*)d_in[6];
  const float* bo1 = (const float*)d_in[7];
  const float* Wo2 = (const float*)d_in[8];
  const float* bo2 = (const float*)d_in[9];
  float* out = (float*)d_out;

  constexpr int N = 512, D = 128, H = 256, O = 256;
  constexpr int NH = N * H;

  float* ws     = (float*)d_ws;      // 5*NH + N floats = ~2.6 MB, L2-resident
  float* pa     = ws;
  float* pb     = ws + 1 * NH;
  float* S      = ws + 2 * NH;
  float* pred   = ws + 3 * NH;
  float* h2     = ws + 4 * NH;
  float* rowsum = ws + 5 * NH;

  const dim3 blk(128);                               // 4 waves/block, 1 tile/wave
  auto grid = [](int M, int Nn) { return dim3(((M / 16) * (Nn / 16)) / 4); };

  // pa = x @ W1[:, :D]^T ; pb = x @ W1[:, D:]^T
  wmma_gemm_wt<<<grid(N, H), blk, 0, stream>>>(x, W1, pa, N, H, D, 2 * D, 0,
                                               nullptr, nullptr, nullptr, 1.0f, 0);
  wmma_gemm_wt<<<grid(N, H), blk, 0, stream>>>(x, W1, pb, N, H, D, 2 * D, D,
                                               nullptr, nullptr, nullptr, 1.0f, 0);
  // S[i,h] = sum_j adj[i,j]*relu(pa[j,h]+pb[i,h]+b1[h]) ; rowsum[i] = sum_j adj[i,j]
  edge_aggregate<<<dim3(N), dim3(H), 0, stream>>>(pa, pb, b1, adj, S, rowsum, N, H);
  // pred = (1/N)*(S @ W2^T) + b2 * rowsum / N
  wmma_gemm_wt<<<grid(N, H), blk, 0, stream>>>(S, W2, pred, N, H, H, H, 0,
                                               nullptr, rowsum, b2, 1.0f / N, 0);
  // h2 = relu(pred @ Wo1^T + bo1)
  wmma_gemm_wt<<<grid(N, H), blk, 0, stream>>>(pred, Wo1, h2, N, H, H, H, 0,
                                               bo1, nullptr, nullptr, 1.0f, 1);
  // out = h2 @ Wo2^T + bo2
  wmma_gemm_wt<<<grid(N, O), blk, 0, stream>>>(h2, Wo2, out, N, O, H, H, 0,
                                               bo2, nullptr, nullptr, 1.0f, 0);
}